// HeteroVGAE_41300405518930
// MI455X (gfx1250) — compile-verified
//
#include <hip/hip_runtime.h>
#include <math.h>

// ---------------------------------------------------------------------------
// HeteroVGAE forward for MI455X (gfx1250, wave32).
//   scatter (L2-resident f32 atomics) ->
//   h = (summed@W_l)*inv_cnt + b_l + x_gene@W_r      [WMMA f32 16x16x4]
//   mu/logstd = h@{mu_W,lv_W}+bias                   [WMMA f32 16x16x4]
//   z = mu + eps*exp(logstd)
// Block = 256 threads (8 wave32s) owning a 64-gene x 128-col tile:
// 4 row tiles share each B (weight) fragment -> 8 WMMA per fragment load set.
// ---------------------------------------------------------------------------

typedef float v2f __attribute__((ext_vector_type(2)));
typedef float v8f __attribute__((ext_vector_type(8)));

#define WMMA_F32_16X16X4(a, b, c) \
  __builtin_amdgcn_wmma_f32_16x16x4_f32(false, (a), false, (b), (short)0, (c), false, false)

#define DDIM 128
#define LDS_STRIDE 132  // 128 + 4 pad: conflict-free stores/loads, rows 16B-aligned
#define RTILES 4        // 16-row tiles per block -> 64 genes/block

// --- zero the workspace accumulators (summed + cnt) ------------------------
__global__ void zero_f32_kernel(float* __restrict__ p, size_t n) {
  size_t i = (size_t)blockIdx.x * blockDim.x + threadIdx.x;
  const size_t stride = (size_t)gridDim.x * blockDim.x;
  for (; i < n; i += stride) p[i] = 0.0f;
}

// --- edge scatter: one wave per edge, float4 per lane ----------------------
__global__ void __launch_bounds__(256)
sage_scatter_kernel(const float* __restrict__ x_disease,
                    const int* __restrict__ edge_src,
                    const int* __restrict__ edge_dst,
                    float* __restrict__ summed,
                    float* __restrict__ cnt,
                    int n_edges) {
  const int lane = threadIdx.x & 31;
  const int wave = (blockIdx.x << 3) + (threadIdx.x >> 5);
  if (wave >= n_edges) return;
  const int s = edge_src[wave];
  const int d = edge_dst[wave];
  // coalesced 512B row read; x_disease (10MB) is L2-resident
  const float4 v = *(const float4*)(x_disease + (size_t)s * DDIM + lane * 4);
  float* out = summed + (size_t)d * DDIM + lane * 4;
  // no-return agent-scope f32 atomics -> HW global_atomic_add_f32 at L2
  __hip_atomic_fetch_add(out + 0, v.x, __ATOMIC_RELAXED, __HIP_MEMORY_SCOPE_AGENT);
  __hip_atomic_fetch_add(out + 1, v.y, __ATOMIC_RELAXED, __HIP_MEMORY_SCOPE_AGENT);
  __hip_atomic_fetch_add(out + 2, v.z, __ATOMIC_RELAXED, __HIP_MEMORY_SCOPE_AGENT);
  __hip_atomic_fetch_add(out + 3, v.w, __ATOMIC_RELAXED, __HIP_MEMORY_SCOPE_AGENT);
  if (lane == 0)
    __hip_atomic_fetch_add(cnt + d, 1.0f, __ATOMIC_RELAXED, __HIP_MEMORY_SCOPE_AGENT);
}

// --- fused SAGE linear + VGAE heads, WMMA f32 16x16x4 ----------------------
__global__ void __launch_bounds__(256)
vgae_fused_kernel(const float* __restrict__ summed,
                  const float* __restrict__ cnt,
                  const float* __restrict__ x_gene,
                  const float* __restrict__ eps,
                  const float* __restrict__ W_l,
                  const float* __restrict__ b_l,
                  const float* __restrict__ W_r,
                  const float* __restrict__ mu_W,
                  const float* __restrict__ mu_b,
                  const float* __restrict__ lv_W,
                  const float* __restrict__ lv_b,
                  float* __restrict__ z,
                  int n_gene) {
  __shared__ float h_lds[16 * RTILES * LDS_STRIDE];

  const int lane = threadIdx.x & 31;
  const int wv   = threadIdx.x >> 5;   // 0..7 -> column tile
  const int hi   = lane >> 4;          // 0: K pair {0,1}, 1: K pair {2,3}
  const int m    = lane & 15;          // A row / C col index within tile
  const int n    = wv * 16 + m;        // output column 0..127
  const int g0   = blockIdx.x * (16 * RTILES);

  // clamped A-row bases (EXEC must stay all-1s through WMMA)
  const float* srow[RTILES];
  const float* xrow[RTILES];
#pragma unroll
  for (int t = 0; t < RTILES; ++t) {
    const int gm = min(g0 + 16 * t + m, n_gene - 1);
    srow[t] = summed + (size_t)gm * DDIM;
    xrow[t] = x_gene + (size_t)gm * DDIM;
  }

  // ---- stage 1: accS = summed@W_l ; accX = x_gene@W_r (separate accs so the
  //      1/cnt row scaling happens once in the epilogue, not per k-step) ----
  v8f accS[RTILES], accX[RTILES];
#pragma unroll
  for (int t = 0; t < RTILES; ++t) {
    accS[t] = (v8f){0.f, 0.f, 0.f, 0.f, 0.f, 0.f, 0.f, 0.f};
    accX[t] = (v8f){0.f, 0.f, 0.f, 0.f, 0.f, 0.f, 0.f, 0.f};
  }

  for (int k0 = 0; k0 < DDIM; k0 += 4) {
    const int ka = k0 + 2 * hi;  // this lane-half supplies K = ka, ka+1
    // B fragments (4x16 f32): shared by all 4 row tiles -> 8 WMMAs per load set
    v2f bl; bl[0] = W_l[(size_t)ka * DDIM + n]; bl[1] = W_l[(size_t)(ka + 1) * DDIM + n];
    v2f br; br[0] = W_r[(size_t)ka * DDIM + n]; br[1] = W_r[(size_t)(ka + 1) * DDIM + n];
#pragma unroll
    for (int t = 0; t < RTILES; ++t) {
      const float2 sv = *(const float2*)(srow[t] + ka);
      const float2 xv = *(const float2*)(xrow[t] + ka);
      v2f a_s; a_s[0] = sv.x; a_s[1] = sv.y;
      v2f a_x; a_x[0] = xv.x; a_x[1] = xv.y;
      accS[t] = WMMA_F32_16X16X4(a_s, bl, accS[t]);
      accX[t] = WMMA_F32_16X16X4(a_x, br, accX[t]);
    }
  }

  const float bias_l = b_l[n];
#pragma unroll
  for (int t = 0; t < RTILES; ++t) {
#pragma unroll
    for (int r = 0; r < 8; ++r) {
      // C/D layout: VGPR r, lanes 0-15 -> row r; lanes 16-31 -> row r+8
      const int row = 16 * t + r + 8 * hi;
      const int g   = min(g0 + row, n_gene - 1);
      const float inv = 1.0f / fmaxf(cnt[g], 1.0f);
      h_lds[row * LDS_STRIDE + n] = accS[t][r] * inv + accX[t][r] + bias_l;
    }
  }
  __syncthreads();

  // ---- stage 2: mu / logstd heads from LDS-resident h ----
  v8f amu[RTILES], alv[RTILES];
#pragma unroll
  for (int t = 0; t < RTILES; ++t) {
    amu[t] = (v8f){0.f, 0.f, 0.f, 0.f, 0.f, 0.f, 0.f, 0.f};
    alv[t] = (v8f){0.f, 0.f, 0.f, 0.f, 0.f, 0.f, 0.f, 0.f};
  }

  for (int k0 = 0; k0 < DDIM; k0 += 4) {
    const int ka = k0 + 2 * hi;
    v2f bm; bm[0] = mu_W[(size_t)ka * DDIM + n]; bm[1] = mu_W[(size_t)(ka + 1) * DDIM + n];
    v2f bv; bv[0] = lv_W[(size_t)ka * DDIM + n]; bv[1] = lv_W[(size_t)(ka + 1) * DDIM + n];
#pragma unroll
    for (int t = 0; t < RTILES; ++t) {
      // rows are 16B-aligned (132*4B) and ka is even -> 8B-aligned ds_load_b64
      const float2 hv = *(const float2*)(&h_lds[(16 * t + m) * LDS_STRIDE + ka]);
      v2f ah; ah[0] = hv.x; ah[1] = hv.y;
      amu[t] = WMMA_F32_16X16X4(ah, bm, amu[t]);
      alv[t] = WMMA_F32_16X16X4(ah, bv, alv[t]);
    }
  }

  const float mb = mu_b[n];
  const float vb = lv_b[n];
#pragma unroll
  for (int t = 0; t < RTILES; ++t) {
#pragma unroll
    for (int r = 0; r < 8; ++r) {
      const int g = g0 + 16 * t + r + 8 * hi;
      if (g < n_gene) {
        const float mu  = amu[t][r] + mb;
        const float lsd = alv[t][r] + vb;
        const size_t idx = (size_t)g * DDIM + n;
        z[idx] = mu + eps[idx] * expf(lsd);
      }
    }
  }
}

// ---------------------------------------------------------------------------
extern "C" void kernel_launch(void* const* d_in, const int* in_sizes, int n_in,
                              void* d_out, int out_size, void* d_ws, size_t ws_size,
                              hipStream_t stream) {
  (void)n_in; (void)out_size; (void)ws_size;

  const float* x_disease = (const float*)d_in[0];
  const float* x_gene    = (const float*)d_in[1];
  const int*   edge_src  = (const int*)d_in[2];
  const int*   edge_dst  = (const int*)d_in[3];
  const float* eps       = (const float*)d_in[4];
  const float* W_l       = (const float*)d_in[5];
  const float* b_l       = (const float*)d_in[6];
  const float* W_r       = (const float*)d_in[7];
  const float* mu_W      = (const float*)d_in[8];
  const float* mu_b      = (const float*)d_in[9];
  const float* lv_W      = (const float*)d_in[10];
  const float* lv_b      = (const float*)d_in[11];
  float* z = (float*)d_out;

  const int n_gene  = in_sizes[1] / DDIM;
  const int n_edges = in_sizes[2];

  float* summed = (float*)d_ws;                       // n_gene * 128 floats
  float* cnt    = summed + (size_t)n_gene * DDIM;     // n_gene floats

  {  // zero accumulators (workspace is poisoned / persists between replays)
    const size_t tot = (size_t)n_gene * (DDIM + 1);
    int blocks = (int)((tot + 255) / 256);
    if (blocks > 16384) blocks = 16384;
    zero_f32_kernel<<<blocks, 256, 0, stream>>>(summed, tot);
  }
  {  // one wave per edge, 8 waves per block
    const int blocks = (n_edges + 7) / 8;
    sage_scatter_kernel<<<blocks, 256, 0, stream>>>(x_disease, edge_src, edge_dst,
                                                    summed, cnt, n_edges);
  }
  {  // 64 genes x 128 cols per block
    const int blocks = (n_gene + (16 * RTILES) - 1) / (16 * RTILES);
    vgae_fused_kernel<<<blocks, 256, 0, stream>>>(summed, cnt, x_gene, eps,
                                                  W_l, b_l, W_r, mu_W, mu_b,
                                                  lv_W, lv_b, z, n_gene);
  }
}